// SNNReadoutOnly_57380763075010
// MI455X (gfx1250) — compile-verified
//
#include <hip/hip_runtime.h>
#include <hip/hip_bf16.h>

// ---------------------------------------------------------------------------
// SNN readout: I = x@W1+b1 ; 200-step Izhikevich ; center ; @W2+b2
// B=2048, Din=1024, H=4096, O=10, T=200  (all fp32)
// ---------------------------------------------------------------------------

typedef __attribute__((ext_vector_type(2))) float v2f;
typedef __attribute__((ext_vector_type(8))) float v8f;

#define DIM_B   2048
#define DIM_K   1024
#define DIM_H   4096
#define DIM_O   10
#define T_STEPS 200

#define BM 32          // block rows (batch)
#define BN 64          // block cols (hidden)
#define KC 32          // K chunk
#define SA_STRIDE 36   // padded, keeps 16B row alignment, 8B frag alignment
#define SB_STRIDE 68   // padded, keeps 16B row alignment

// ---------------------------------------------------------------------------
// CDNA5 async copy global->LDS (ASYNCcnt-tracked, no VGPR round trip).
// Low 32 bits of a generic pointer to __shared__ are the LDS byte offset
// (ISA 10.2).
// ---------------------------------------------------------------------------
__device__ __forceinline__ void async_copy_b128(void* lds_ptr, const void* gptr)
{
    const unsigned lds_off = (unsigned)(uintptr_t)lds_ptr;
    asm volatile("global_load_async_to_lds_b128 %0, %1, off"
                 :: "v"(lds_off), "v"(gptr) : "memory");
}
__device__ __forceinline__ void wait_async_zero()
{
    asm volatile("s_wait_asynccnt 0x0" ::: "memory");
}

// ---------------------------------------------------------------------------
// Kernel A: fp32 WMMA GEMM (I = x@W1 + b1) fused with Izhikevich simulation.
// 256 threads = 8 waves; each wave owns one 16x16 tile of the 32x64 block
// tile. Double-buffered LDS filled by global_load_async_to_lds_b128.
// ---------------------------------------------------------------------------
__global__ __launch_bounds__(256) void snn_gemm_izh(
    const float* __restrict__ x,    // (2048,1024) row-major
    const float* __restrict__ W1,   // (1024,4096) row-major
    const float* __restrict__ b1,   // (4096)
    float* __restrict__ S)          // (2048,4096) spike counts out
{
    __shared__ float sA[2][BM * SA_STRIDE];
    __shared__ float sB[2][KC * SB_STRIDE];

    const int tid   = threadIdx.x;
    const int lane  = tid & 31;
    const int wave  = tid >> 5;
    const int gRow0 = blockIdx.y * BM;
    const int gCol0 = blockIdx.x * BN;

    const int rsub  = (wave >> 2) << 4;   // 0 or 16 : wave row sub-tile
    const int csub  = (wave & 3) << 4;    // 0,16,32,48 : wave col sub-tile
    const int lhalf = lane >> 4;          // 0: lanes 0-15, 1: lanes 16-31
    const int lmod  = lane & 15;

    // cooperative-copy coordinates (one float4 of x, two float4 of W1/thread)
    const int aRow = (tid * 4) >> 5;
    const int aK   = (tid * 4) & 31;
    const int bKr0 = (tid * 4) >> 6;
    const int bCc0 = (tid * 4) & 63;
    const int bKr1 = ((tid + 256) * 4) >> 6;
    const int bCc1 = ((tid + 256) * 4) & 63;

    v8f acc = {0.f, 0.f, 0.f, 0.f, 0.f, 0.f, 0.f, 0.f};

    // issue one K-chunk's 3 async B128 copies into buffer p
    auto issue_chunk = [&](int k0, int p) {
        async_copy_b128(&sA[p][aRow * SA_STRIDE + aK],
                        x + (size_t)(gRow0 + aRow) * DIM_K + k0 + aK);
        async_copy_b128(&sB[p][bKr0 * SB_STRIDE + bCc0],
                        W1 + (size_t)(k0 + bKr0) * DIM_H + gCol0 + bCc0);
        async_copy_b128(&sB[p][bKr1 * SB_STRIDE + bCc1],
                        W1 + (size_t)(k0 + bKr1) * DIM_H + gCol0 + bCc1);
    };

    issue_chunk(0, 0);
    int p = 0;
    for (int k0 = 0; k0 < DIM_K; k0 += KC) {
        wait_async_zero();     // this chunk's copies landed in LDS
        __syncthreads();       // ...for every wave; spare buffer free to reuse

        if (k0 + KC < DIM_K)
            issue_chunk(k0 + KC, p ^ 1);     // overlaps with WMMAs below
        if (k0 + 2 * KC < DIM_K)             // warm L2 one more chunk ahead
            __builtin_prefetch(W1 + (size_t)(k0 + 2 * KC) * DIM_H + gCol0 + (tid & 63), 0, 3);

        // ---- 8 x v_wmma_f32_16x16x4_f32 over this K chunk
        #pragma unroll
        for (int kk = 0; kk < KC; kk += 4) {
            const int kb = kk + 2 * lhalf;  // lanes 16-31 carry K+2,K+3 (ISA 7.12.2)
            // A fragment: lane holds row (M=lmod), 2 consecutive K in v[0],v[1]
            v2f a = *(const v2f*)(&sA[p][(rsub + lmod) * SA_STRIDE + kb]);
            // B fragment: lane holds col (N=lmod), K rows kb, kb+1 in v[0],v[1]
            v2f b;
            b.x = sB[p][(kb + 0) * SB_STRIDE + csub + lmod];
            b.y = sB[p][(kb + 1) * SB_STRIDE + csub + lmod];
            acc = __builtin_amdgcn_wmma_f32_16x16x4_f32(
                /*neg_a=*/false, a, /*neg_b=*/false, b,
                /*c_mod=*/(short)0, acc, /*reuse_a=*/false, /*reuse_b=*/false);
        }
        p ^= 1;
    }

    // ---- epilogue: bias + 200-step Izhikevich recurrence, all in registers.
    // C/D layout: acc[r] -> row (rsub + r + 8*lhalf), col (csub + lmod)
    const float bias = b1[gCol0 + csub + lmod];
    float Idrv[8], vv[8], uu[8], Sc[8];
    #pragma unroll
    for (int r = 0; r < 8; ++r) {
        Idrv[r] = acc[r] + bias;
        vv[r] = -56.0f;           // C_RESET
        uu[r] = -112.0f;          // B_IZH * C_RESET
        Sc[r] = 0.0f;
    }
    for (int t = 0; t < T_STEPS; ++t) {
        #pragma unroll
        for (int r = 0; r < 8; ++r) {
            const float vo = vv[r];
            const float uo = uu[r];
            const float dv = fmaf(fmaf(0.04f, vo, 5.0f), vo, 140.0f) - uo + Idrv[r];
            const float du = 0.2f * fmaf(2.0f, vo, -uo);
            const float vn = vo + dv;
            const float un = uo + du;
            const bool  sp = (vn >= 30.0f);
            Sc[r] += sp ? 1.0f : 0.0f;
            vv[r]  = sp ? -56.0f : vn;
            uu[r]  = sp ? (un - 13.0f) : un;
        }
    }
    // ---- write spike counts
    const int col = gCol0 + csub + lmod;
    #pragma unroll
    for (int r = 0; r < 8; ++r) {
        const int row = gRow0 + rsub + r + 8 * lhalf;
        S[(size_t)row * DIM_H + col] = Sc[r];
    }
}

// ---------------------------------------------------------------------------
// Kernel B0: zero the column-sum accumulator (ws is poisoned, not zeroed).
// ---------------------------------------------------------------------------
__global__ __launch_bounds__(256) void snn_zero_colsum(float* __restrict__ colsum)
{
    const int i = blockIdx.x * 256 + threadIdx.x;
    if (i < DIM_H) colsum[i] = 0.0f;
}

// ---------------------------------------------------------------------------
// Kernel B1: colsum[h] = sum_b S[b,h]  (partial rows per block + atomicAdd)
// ---------------------------------------------------------------------------
__global__ __launch_bounds__(256) void snn_colsum(
    const float* __restrict__ S, float* __restrict__ colsum)
{
    const int h  = blockIdx.x * 256 + threadIdx.x;   // 16 blocks cover 4096 cols
    const int r0 = blockIdx.y * (DIM_B / 8);         // 8 row chunks of 256 rows
    float s = 0.0f;
    for (int b = 0; b < DIM_B / 8; ++b)
        s += S[(size_t)(r0 + b) * DIM_H + h];
    atomicAdd(&colsum[h], s);
}

// ---------------------------------------------------------------------------
// Kernel C: out[b,o] = (1/T) * sum_h (S[b,h] - colsum[h]/B) * W2[h,o] + b2[o]
// One block per batch row; memory-bound stream of S.
// ---------------------------------------------------------------------------
__global__ __launch_bounds__(256) void snn_readout(
    const float* __restrict__ S, const float* __restrict__ colsum,
    const float* __restrict__ W2,     // (4096,10) row-major
    const float* __restrict__ b2,     // (10)
    float* __restrict__ out)          // (2048,10)
{
    __shared__ float red[256][DIM_O];
    const int b = blockIdx.x;
    const int t = threadIdx.x;

    float p[DIM_O];
    #pragma unroll
    for (int o = 0; o < DIM_O; ++o) p[o] = 0.0f;

    for (int h = t; h < DIM_H; h += 256) {
        const float s = S[(size_t)b * DIM_H + h] - colsum[h] * (1.0f / (float)DIM_B);
        const float* w = W2 + (size_t)h * DIM_O;
        #pragma unroll
        for (int o = 0; o < DIM_O; ++o) p[o] = fmaf(s, w[o], p[o]);
    }
    #pragma unroll
    for (int o = 0; o < DIM_O; ++o) red[t][o] = p[o];
    __syncthreads();
    for (int st = 128; st > 0; st >>= 1) {
        if (t < st) {
            #pragma unroll
            for (int o = 0; o < DIM_O; ++o) red[t][o] += red[t + st][o];
        }
        __syncthreads();
    }
    if (t < DIM_O)
        out[(size_t)b * DIM_O + t] = red[0][t] * (1.0f / (float)T_STEPS) + b2[t];
}

// ---------------------------------------------------------------------------
extern "C" void kernel_launch(void* const* d_in, const int* in_sizes, int n_in,
                              void* d_out, int out_size, void* d_ws, size_t ws_size,
                              hipStream_t stream)
{
    const float* x  = (const float*)d_in[0];   // (2048,1024)
    const float* W1 = (const float*)d_in[1];   // (1024,4096)
    const float* b1 = (const float*)d_in[2];   // (4096)
    const float* W2 = (const float*)d_in[3];   // (4096,10)
    const float* b2 = (const float*)d_in[4];   // (10)
    float* out = (float*)d_out;                // (2048,10)

    float* S      = (float*)d_ws;                       // 2048*4096 floats
    float* colsum = S + (size_t)DIM_B * DIM_H;          // 4096 floats

    dim3 gridA(DIM_H / BN, DIM_B / BM);                 // (64, 64)
    snn_gemm_izh<<<gridA, 256, 0, stream>>>(x, W1, b1, S);

    snn_zero_colsum<<<DIM_H / 256, 256, 0, stream>>>(colsum);
    snn_colsum<<<dim3(DIM_H / 256, 8), 256, 0, stream>>>(S, colsum);
    snn_readout<<<DIM_B, 256, 0, stream>>>(S, colsum, W2, b2, out);
}